// StateSpaceLayer_13151189860393
// MI455X (gfx1250) — compile-verified
//
#include <hip/hip_runtime.h>

typedef __attribute__((ext_vector_type(2))) float v2f;
typedef __attribute__((ext_vector_type(8))) float v8f;

#define D_MODEL 1024
#define BATCH   8
#define SEQ     2048
#define NROWS   (BATCH * SEQ)        // 16384 rows of the [rows, D] GEMMs

#define BM 128                       // block tile M
#define BN 128                       // block tile N
#define KC 32                        // K chunk staged in LDS
#define LDSTR (KC + 4)               // +4 floats pad: 144B rows -> 16B aligned, bank-conflict free
#define NCHUNK (D_MODEL / KC)

static_assert(NROWS % BM == 0, "M tiling");
static_assert(D_MODEL % BN == 0, "N tiling");
static_assert(D_MODEL % KC == 0, "K tiling");

enum { EP_BIAS = 0, EP_GATE = 1, EP_RESID = 2 };

// Issue one 128 x KC fp32 tile as 4 async b128 loads per lane (1024 total per block).
// Data lands directly in LDS; completion tracked by ASYNCcnt.
__device__ __forceinline__ void async_load_tile(const float* __restrict__ gbase, // + row0*ldg + kc
                                                float (*lds)[LDSTR], int tid)
{
#pragma unroll
    for (int t = 0; t < 4; ++t) {
        int idx = tid + t * 256;            // 0..1023
        int row = idx >> 3, kq = idx & 7;   // 8 float4 per row
        const float* gp = gbase + (size_t)row * D_MODEL + kq * 4;
        uint32_t lp = (uint32_t)(uintptr_t)&lds[row][kq * 4];   // low 32 bits = LDS byte addr
        asm volatile("global_load_async_to_lds_b128 %0, %1, off"
                     :: "v"(lp), "v"(gp) : "memory");
    }
}

__device__ __forceinline__ void wait_async0()
{
#if __has_builtin(__builtin_amdgcn_s_wait_asynccnt)
    __builtin_amdgcn_s_wait_asynccnt(0);
#else
    asm volatile("s_wait_asynccnt 0x0" ::: "memory");
#endif
}

// C[M,N] = X[M,K] @ W[N,K]^T (+ epilogue), via V_WMMA_F32_16X16X4_F32.
// Grid: (N/BN, M/BM), block: 256 threads = 8 waves (wave32) in a 4(M) x 2(N) grid.
// Each wave: 2 M-tiles x 4 N-tiles = 8 16x16 accumulators (64 VGPRs).
// LDS tiles are double-buffered and filled with async-to-LDS loads overlapped with WMMA.
template <int EP>
__global__ __launch_bounds__(256)
void gemm_wmma_f32(const float* __restrict__ Xg,
                   const float* __restrict__ Wg,    // [N][K] row-major (acts as B^T)
                   const float* __restrict__ bias,  // [N]
                   const float* __restrict__ aux,   // EP_GATE: A decay [N]; EP_RESID: Dp [N]
                   const float* __restrict__ res,   // EP_RESID: residual x [M][N]
                   float* __restrict__ Og)
{
    __shared__ float Xs[2][BM][LDSTR];   // 2 x 18 KB
    __shared__ float Ws[2][BN][LDSTR];   // 2 x 18 KB

    const int tid     = threadIdx.x;
    const int wid     = tid >> 5;        // wave id 0..7 (wave32)
    const int lane    = tid & 31;
    const int lanelow = lane & 15;
    const int half    = lane >> 4;       // 0: lanes 0-15, 1: lanes 16-31

    const int m0 = blockIdx.y * BM;
    const int n0 = blockIdx.x * BN;
    const int wm = (wid >> 1) * 32;      // wave's M offset inside tile (2 m-tiles)
    const int wn = (wid & 1) * 64;       // wave's N offset inside tile (4 n-tiles)

    const float* Xbase = Xg + (size_t)m0 * D_MODEL;
    const float* Wbase = Wg + (size_t)n0 * D_MODEL;

    v8f zero = {};
    v8f acc[2][4];
#pragma unroll
    for (int mt = 0; mt < 2; ++mt)
#pragma unroll
        for (int j = 0; j < 4; ++j) acc[mt][j] = zero;

    // prologue: fill buffer 0
    async_load_tile(Xbase, Xs[0], tid);
    async_load_tile(Wbase, Ws[0], tid);

    for (int c = 0; c < NCHUNK; ++c) {
        const int buf = c & 1;
        wait_async0();        // own async writes for chunk c are in LDS
        __syncthreads();      // everyone's writes visible; everyone done reading buf^1

        if (c + 1 < NCHUNK) { // prefetch chunk c+1 into the other buffer, hidden under WMMAs
            const int kc2 = (c + 1) * KC;
            async_load_tile(Xbase + kc2, Xs[buf ^ 1], tid);
            async_load_tile(Wbase + kc2, Ws[buf ^ 1], tid);
        }

        const float (*xs)[LDSTR] = Xs[buf];
        const float (*ws)[LDSTR] = Ws[buf];

#pragma unroll
        for (int ks = 0; ks < KC; ks += 4) {
            // A 16x4 layout: VGPR0 = K{0|2}, VGPR1 = K{1|3} per lane half (B mirrored)
            int kk = ks + 2 * half;
            v2f a0 = *(const v2f*)&xs[wm + lanelow][kk];
            v2f a1 = *(const v2f*)&xs[wm + 16 + lanelow][kk];
#pragma unroll
            for (int j = 0; j < 4; ++j) {
                v2f bf = *(const v2f*)&ws[wn + 16 * j + lanelow][kk];
                acc[0][j] = __builtin_amdgcn_wmma_f32_16x16x4_f32(
                    false, a0, false, bf, (short)0, acc[0][j], false, false);
                acc[1][j] = __builtin_amdgcn_wmma_f32_16x16x4_f32(
                    false, a1, false, bf, (short)0, acc[1][j], false, false);
            }
        }
        // no trailing barrier needed: next iteration's wait+barrier orders reuse
    }

    // --- epilogue: C layout is VGPR r -> M = r + 8*half, N = lanelow ---
#pragma unroll
    for (int mt = 0; mt < 2; ++mt) {
#pragma unroll
        for (int j = 0; j < 4; ++j) {
            int col = n0 + wn + 16 * j + lanelow;
            float bv = bias[col];
            float av = (EP == EP_BIAS) ? 0.0f : aux[col];
#pragma unroll
            for (int r = 0; r < 8; ++r) {
                int row = m0 + wm + 16 * mt + 8 * half + r;
                float v = acc[mt][j][r] + bv;
                if (EP == EP_GATE) {
                    float sg = 1.0f / (1.0f + expf(-v));   // sigmoid
                    v = expf(av * sg);                      // exp(A * sigmoid(lin))
                } else if (EP == EP_RESID) {
                    v += av + res[(size_t)row * D_MODEL + col];
                }
                Og[(size_t)row * D_MODEL + col] = v;
            }
        }
    }
}

// Linear recurrence h_t = a_t * h_{t-1} + bx_t over S, one thread per (b, d) channel.
// 8192 channels; loads/stores coalesced along d. Bandwidth-bound: ~192 MB @ 23.3 TB/s.
__global__ __launch_bounds__(256)
void scan_kernel(const float* __restrict__ Ag,
                 const float* __restrict__ Bx,
                 float* __restrict__ Hs)
{
    int ch = blockIdx.x * 256 + threadIdx.x;     // 0 .. BATCH*D_MODEL-1
    int b = ch >> 10;                            // / D_MODEL
    int d = ch & (D_MODEL - 1);
    size_t base = (size_t)b * SEQ * D_MODEL + d;
    float h = 0.0f;
    for (int s = 0; s < SEQ; ++s) {
        size_t off = base + (size_t)s * D_MODEL;
        h = fmaf(Ag[off], h, Bx[off]);
        Hs[off] = h;
    }
}

extern "C" void kernel_launch(void* const* d_in, const int* in_sizes, int n_in,
                              void* d_out, int out_size, void* d_ws, size_t ws_size,
                              hipStream_t stream)
{
    (void)in_sizes; (void)n_in; (void)out_size; (void)ws_size;
    const float* x  = (const float*)d_in[0];
    const float* A  = (const float*)d_in[1];
    const float* Wb = (const float*)d_in[2];
    const float* bb = (const float*)d_in[3];
    const float* Wc = (const float*)d_in[4];
    const float* bc = (const float*)d_in[5];
    const float* Dp = (const float*)d_in[6];
    const float* Wa = (const float*)d_in[7];
    const float* ba = (const float*)d_in[8];
    const float* Wp = (const float*)d_in[9];
    const float* bp = (const float*)d_in[10];
    float* out = (float*)d_out;

    const size_t elems = (size_t)NROWS * D_MODEL;   // 16M floats = 64 MB per buffer
    float* Ag = (float*)d_ws;          // ws[0 .. 64MB)
    float* Bx = Ag + elems;            // ws[64 .. 128MB)
    float* Hs = Bx + elems;            // ws[128 .. 192MB)
    float* Y  = Ag;                    // Ag is dead after the scan -> reuse for y

    dim3 grid(D_MODEL / BN, NROWS / BM);   // (8, 128)
    dim3 blk(256);

    // 1) A_gate = exp(A * sigmoid(x @ Wa^T + ba))
    gemm_wmma_f32<EP_GATE><<<grid, blk, 0, stream>>>(x, Wa, ba, A, nullptr, Ag);
    // 2) Bx = x @ Wb^T + bb
    gemm_wmma_f32<EP_BIAS><<<grid, blk, 0, stream>>>(x, Wb, bb, nullptr, nullptr, Bx);
    // 3) recurrence over time
    scan_kernel<<<(BATCH * D_MODEL) / 256, blk, 0, stream>>>(Ag, Bx, Hs);
    // 4) y = Hs @ Wc^T + bc + Dp + x
    gemm_wmma_f32<EP_RESID><<<grid, blk, 0, stream>>>(Hs, Wc, bc, Dp, x, Y);
    // 5) out = y @ Wp^T + bp
    gemm_wmma_f32<EP_BIAS><<<grid, blk, 0, stream>>>(Y, Wp, bp, nullptr, nullptr, out);
}